// HNM_27152783245914
// MI455X (gfx1250) — compile-verified
//
#include <hip/hip_runtime.h>
#include <math.h>

#define NROWS 1000000
#define WD 20
#define EPSF 1e-16f

typedef float v2f __attribute__((ext_vector_type(2)));
typedef float v8f __attribute__((ext_vector_type(8)));

// ---- ctrl layout (floats in d_ws) ----
// [0..19] k_r  [20] beta_r [21] g_r [22..24] s_r [25] gamma_r [26] kenorm_r
// [32..51] k_w [52] beta_w [53] g_w [54..56] s_w [57] gamma_w [58] kenorm_w
// [64..83] erase  [84..103] add_pre  [104] rho [105..106] alu_head
// [112..131] add_final
// [160] max_r [161] max_w [162] sum_r [163] sum_w [164] wsum_r [165] wsum_w

__device__ __forceinline__ float sigmoidf_(float x) { return 1.0f / (1.0f + expf(-x)); }
__device__ __forceinline__ float softplusf_(float x) { return log1pf(expf(x)); }

// ------------------------- K0: controller MLP -------------------------
__global__ __launch_bounds__(128) void controller_kernel(
    const float* __restrict__ X,
    const float* __restrict__ W1, const float* __restrict__ b1,
    const float* __restrict__ W2, const float* __restrict__ b2,
    const float* __restrict__ Wxi, const float* __restrict__ bxi,
    const float* __restrict__ Wz, const float* __restrict__ bz,
    float* __restrict__ ctrl)
{
    __shared__ float h1[48], h2[72], xi[92], zeta[3];
    const int tid = threadIdx.x;
    if (tid < 48) {
        float a = b1[tid];
        for (int j = 0; j < 14; ++j) a += X[j] * W1[tid * 14 + j];
        h1[tid] = a;                       // no activation (faithful)
    }
    __syncthreads();
    if (tid < 72) {
        float a = b2[tid];
        for (int j = 0; j < 48; ++j) a += h1[j] * W2[tid * 48 + j];
        h2[tid] = a;
    }
    __syncthreads();
    if (tid < 92) {
        float a = bxi[tid];
        for (int j = 0; j < 72; ++j) a += h2[j] * Wxi[tid * 72 + j];
        xi[tid] = a;
    }
    if (tid >= 92 && tid < 95) {
        int o = tid - 92;
        float a = bz[o];
        for (int j = 0; j < 72; ++j) a += h2[j] * Wz[o * 72 + j];
        zeta[o] = a;
    }
    __syncthreads();
    if (tid < 20) {
        ctrl[tid]      = tanhf(xi[tid]);          // k_r
        ctrl[32 + tid] = tanhf(xi[26 + tid]);     // k_w
        ctrl[64 + tid] = sigmoidf_(xi[52 + tid]); // erase
        ctrl[84 + tid] = tanhf(xi[72 + tid]);     // add_pre
    }
    __syncthreads();
    if (tid == 0) {
        // read head scalars
        ctrl[21] = sigmoidf_(xi[20]);
        float m = fmaxf(xi[21], fmaxf(xi[22], xi[23]));
        float e0 = expf(xi[21] - m), e1 = expf(xi[22] - m), e2 = expf(xi[23] - m);
        float s = e0 + e1 + e2;
        ctrl[22] = e0 / s; ctrl[23] = e1 / s; ctrl[24] = e2 / s;
        ctrl[25] = 1.0f + softplusf_(xi[24]);
        ctrl[20] = softplusf_(xi[25]);
        // write head scalars
        ctrl[53] = sigmoidf_(xi[46]);
        m = fmaxf(xi[47], fmaxf(xi[48], xi[49]));
        e0 = expf(xi[47] - m); e1 = expf(xi[48] - m); e2 = expf(xi[49] - m);
        s = e0 + e1 + e2;
        ctrl[54] = e0 / s; ctrl[55] = e1 / s; ctrl[56] = e2 / s;
        ctrl[57] = 1.0f + softplusf_(xi[50]);
        ctrl[52] = softplusf_(xi[51]);
        // key norms (with EPS like reference)
        float nr = 0.f, nw = 0.f;
        for (int j = 0; j < WD; ++j) {
            float kr = ctrl[j] + EPSF, kw = ctrl[32 + j] + EPSF;
            nr += kr * kr; nw += kw * kw;
        }
        ctrl[26] = fmaxf(sqrtf(nr), 1e-8f);
        ctrl[58] = fmaxf(sqrtf(nw), 1e-8f);
        // rho + alu head
        ctrl[104] = sigmoidf_(zeta[0]);
        float zm = fmaxf(zeta[1], zeta[2]);
        float z0 = expf(zeta[1] - zm), z1 = expf(zeta[2] - zm);
        ctrl[105] = z0 / (z0 + z1);
        ctrl[106] = z1 / (z0 + z1);
    }
}

// ------------------------- K1: WMMA score pass -------------------------
// Computes sim_r/sim_w = beta * cosine(Memory+eps, k+eps); per-block max partials.
__global__ __launch_bounds__(256) void score_kernel(
    const float* __restrict__ Mem, const float* __restrict__ ctrl,
    float* __restrict__ sim_r, float* __restrict__ sim_w,
    float* __restrict__ pmax_r, float* __restrict__ pmax_w)
{
    __shared__ float tile[8][320];   // per-wave: 16 rows x 20 f32 (Me = Mem+eps)
    __shared__ float stage[8][32];   // per-wave: num_r[16], num_w[16]
    __shared__ float red[256];

    const int tid = threadIdx.x;
    const int wave = tid >> 5;
    const int lane = tid & 31;

    const float beta_r = ctrl[20], beta_w = ctrl[52];
    const float ken_r = ctrl[26], ken_w = ctrl[58];

    // Build B fragments (4x16, cols 0/1 = k_r+eps / k_w+eps), per ISA layout:
    // lanes 0-15 hold K pair {0,1}, lanes 16-31 hold K pair {2,3}.
    v2f bfrag[5];
    {
        const int n = lane & 15;
        const int kk = (lane >> 4) * 2;
        for (int c = 0; c < 5; ++c) {
            const int k0 = c * 4;
            float b0 = 0.f, b1 = 0.f;
            if (n == 0)      { b0 = ctrl[k0 + kk] + EPSF;      b1 = ctrl[k0 + kk + 1] + EPSF; }
            else if (n == 1) { b0 = ctrl[32 + k0 + kk] + EPSF; b1 = ctrl[32 + k0 + kk + 1] + EPSF; }
            bfrag[c].x = b0; bfrag[c].y = b1;
        }
    }

    float maxr = -3.0e38f, maxw = -3.0e38f;

    const int numTiles = NROWS / 16;
    const int gwave = blockIdx.x * 8 + wave;
    const int wstride = gridDim.x * 8;

    for (int t = gwave; t < numTiles; t += wstride) {
        const int r0 = t * 16;
        const float* src = Mem + (size_t)r0 * WD;
        // Coalesced stage of 320 dwords into LDS (Me = Mem + eps)
        #pragma unroll
        for (int q = 0; q < 10; ++q)
            tile[wave][lane + q * 32] = src[lane + q * 32] + EPSF;
        // Same-wave DS ops are in-order: store->load visibility guaranteed.

        v8f acc = {};
        const int m = lane & 15;
        const int ks = (lane >> 4) * 2;
        #pragma unroll
        for (int c = 0; c < 5; ++c) {
            v2f a;
            a.x = tile[wave][m * WD + c * 4 + ks];
            a.y = tile[wave][m * WD + c * 4 + ks + 1];
            acc = __builtin_amdgcn_wmma_f32_16x16x4_f32(
                false, a, false, bfrag[c], (short)0, acc, false, false);
        }
        // C layout: lane n (0-15) VGPR j = C[j][n]; lane n+16 VGPR j = C[j+8][n].
        if ((lane & 15) < 2) {
            const int col = lane & 15;          // 0 -> num_r, 1 -> num_w
            const int rbase = (lane >> 4) * 8;
            #pragma unroll
            for (int j = 0; j < 8; ++j)
                stage[wave][col * 16 + rbase + j] = acc[j];
        }
        if (lane < 16) {
            float ss = 0.f;
            #pragma unroll
            for (int j = 0; j < WD; ++j) {
                float v = tile[wave][lane * WD + j];
                ss += v * v;
            }
            const float nrm = fmaxf(sqrtf(ss), 1e-8f);
            const float numr = stage[wave][lane];
            const float numw = stage[wave][16 + lane];
            const float sr = (numr / (nrm * ken_r)) * beta_r;
            const float sw = (numw / (nrm * ken_w)) * beta_w;
            sim_r[r0 + lane] = sr;
            sim_w[r0 + lane] = sw;
            maxr = fmaxf(maxr, sr);
            maxw = fmaxf(maxw, sw);
        }
    }

    red[tid] = maxr; __syncthreads();
    for (int s = 128; s > 0; s >>= 1) { if (tid < s) red[tid] = fmaxf(red[tid], red[tid + s]); __syncthreads(); }
    if (tid == 0) pmax_r[blockIdx.x] = red[0];
    __syncthreads();
    red[tid] = maxw; __syncthreads();
    for (int s = 128; s > 0; s >>= 1) { if (tid < s) red[tid] = fmaxf(red[tid], red[tid + s]); __syncthreads(); }
    if (tid == 0) pmax_w[blockIdx.x] = red[0];
}

// ------------------- generic pairwise reduce (1 block) -------------------
__global__ __launch_bounds__(1024) void reduce_pair_kernel(
    const float* __restrict__ pa, const float* __restrict__ pb, int n,
    float* __restrict__ oa, float* __restrict__ ob, int isSum)
{
    __shared__ float red[1024];
    const int tid = threadIdx.x;
    float a = isSum ? 0.f : -3.0e38f;
    for (int i = tid; i < n; i += 1024) a = isSum ? (a + pa[i]) : fmaxf(a, pa[i]);
    red[tid] = a; __syncthreads();
    for (int s = 512; s > 0; s >>= 1) {
        if (tid < s) red[tid] = isSum ? (red[tid] + red[tid + s]) : fmaxf(red[tid], red[tid + s]);
        __syncthreads();
    }
    if (tid == 0) *oa = red[0];
    __syncthreads();
    float b = isSum ? 0.f : -3.0e38f;
    for (int i = tid; i < n; i += 1024) b = isSum ? (b + pb[i]) : fmaxf(b, pb[i]);
    red[tid] = b; __syncthreads();
    for (int s = 512; s > 0; s >>= 1) {
        if (tid < s) red[tid] = isSum ? (red[tid] + red[tid + s]) : fmaxf(red[tid], red[tid + s]);
        __syncthreads();
    }
    if (tid == 0) *ob = red[0];
}

// ------------------------- K3: exp + partial sums -------------------------
__global__ __launch_bounds__(256) void softmax_exp_kernel(
    float* __restrict__ sim_r, float* __restrict__ sim_w,
    const float* __restrict__ ctrl,
    float* __restrict__ psum_r, float* __restrict__ psum_w)
{
    __shared__ float red[256];
    const float mr = ctrl[160], mw = ctrl[161];
    const int tid = threadIdx.x;
    const int gid = blockIdx.x * blockDim.x + tid;
    const int stride = gridDim.x * blockDim.x;
    float sr = 0.f, sw = 0.f;
    for (int i = gid; i < NROWS; i += stride) {
        float er = expf(sim_r[i] - mr);
        float ew = expf(sim_w[i] - mw);
        sim_r[i] = er; sim_w[i] = ew;
        sr += er; sw += ew;
    }
    red[tid] = sr; __syncthreads();
    for (int s = 128; s > 0; s >>= 1) { if (tid < s) red[tid] += red[tid + s]; __syncthreads(); }
    if (tid == 0) psum_r[blockIdx.x] = red[0];
    __syncthreads();
    red[tid] = sw; __syncthreads();
    for (int s = 128; s > 0; s >>= 1) { if (tid < s) red[tid] += red[tid + s]; __syncthreads(); }
    if (tid == 0) psum_w[blockIdx.x] = red[0];
}

// ------------------------- K5: interpolation gate -------------------------
__global__ __launch_bounds__(256) void wg_kernel(
    const float* __restrict__ er, const float* __restrict__ ew,
    const float* __restrict__ rprev, const float* __restrict__ wprev,
    const float* __restrict__ ctrl,
    float* __restrict__ wg_r, float* __restrict__ wg_w)
{
    const float gr = ctrl[21], gw = ctrl[53];
    const float isr = 1.0f / ctrl[162], isw = 1.0f / ctrl[163];
    const int gid = blockIdx.x * blockDim.x + threadIdx.x;
    const int stride = gridDim.x * blockDim.x;
    for (int i = gid; i < NROWS; i += stride) {
        wg_r[i] = gr * (er[i] * isr) + (1.0f - gr) * rprev[i];
        wg_w[i] = gw * (ew[i] * isw) + (1.0f - gw) * wprev[i];
    }
}

// ------------------- K6: circular shift conv + sharpen -------------------
__global__ __launch_bounds__(256) void conv_pow_kernel(
    const float* __restrict__ wg_r, const float* __restrict__ wg_w,
    const float* __restrict__ ctrl,
    float* __restrict__ rw_o, float* __restrict__ ww_o,
    float* __restrict__ pws_r, float* __restrict__ pws_w)
{
    __shared__ float red[256];
    const float sr0 = ctrl[22], sr1 = ctrl[23], sr2 = ctrl[24], gam_r = ctrl[25];
    const float sw0 = ctrl[54], sw1 = ctrl[55], sw2 = ctrl[56], gam_w = ctrl[57];
    const int tid = threadIdx.x;
    const int gid = blockIdx.x * blockDim.x + tid;
    const int stride = gridDim.x * blockDim.x;
    float accr = 0.f, accw = 0.f;
    for (int i = gid; i < NROWS; i += stride) {
        const int im1 = (i == 0) ? (NROWS - 1) : (i - 1);
        const int ip1 = (i == NROWS - 1) ? 0 : (i + 1);
        float vr = sr0 * wg_r[im1] + sr1 * wg_r[i] + sr2 * wg_r[ip1];
        float vw = sw0 * wg_w[im1] + sw1 * wg_w[i] + sw2 * wg_w[ip1];
        vr = powf(vr, gam_r);
        vw = powf(vw, gam_w);
        rw_o[i] = vr; ww_o[i] = vw;
        accr += vr; accw += vw;
    }
    red[tid] = accr; __syncthreads();
    for (int s = 128; s > 0; s >>= 1) { if (tid < s) red[tid] += red[tid + s]; __syncthreads(); }
    if (tid == 0) pws_r[blockIdx.x] = red[0];
    __syncthreads();
    red[tid] = accw; __syncthreads();
    for (int s = 128; s > 0; s >>= 1) { if (tid < s) red[tid] += red[tid + s]; __syncthreads(); }
    if (tid == 0) pws_w[blockIdx.x] = red[0];
}

// ------------------------- K8: normalize weights -------------------------
__global__ __launch_bounds__(256) void norm_kernel(
    float* __restrict__ rw_o, float* __restrict__ ww_o, const float* __restrict__ ctrl)
{
    const float ir = 1.0f / (ctrl[164] + EPSF);
    const float iw = 1.0f / (ctrl[165] + EPSF);
    const int gid = blockIdx.x * blockDim.x + threadIdx.x;
    const int stride = gridDim.x * blockDim.x;
    for (int i = gid; i < NROWS; i += stride) {
        rw_o[i] *= ir;
        ww_o[i] *= iw;
    }
}

// ------------------- K9: new_read_head partials (rw @ Memory) -------------------
__global__ __launch_bounds__(256) void readvec_kernel(
    const float* __restrict__ Mem, const float* __restrict__ rw_o,
    float* __restrict__ pnrh)
{
    __shared__ float red[256];
    float acc[WD];
    #pragma unroll
    for (int j = 0; j < WD; ++j) acc[j] = 0.f;
    const int tid = threadIdx.x;
    const int gid = blockIdx.x * blockDim.x + tid;
    const int stride = gridDim.x * blockDim.x;
    for (int i = gid; i < NROWS; i += stride) {
        const float wv = rw_o[i];
        const float4* row = (const float4*)(Mem + (size_t)i * WD);
        #pragma unroll
        for (int q = 0; q < 5; ++q) {
            float4 m = row[q];
            acc[q * 4 + 0] += wv * m.x;
            acc[q * 4 + 1] += wv * m.y;
            acc[q * 4 + 2] += wv * m.z;
            acc[q * 4 + 3] += wv * m.w;
        }
    }
    for (int j = 0; j < WD; ++j) {
        red[tid] = acc[j]; __syncthreads();
        for (int s = 128; s > 0; s >>= 1) { if (tid < s) red[tid] += red[tid + s]; __syncthreads(); }
        if (tid == 0) pnrh[blockIdx.x * WD + j] = red[0];
        __syncthreads();
    }
}

// ------------------------- K10: ALU nets + output head -------------------------
__device__ void dense_sh(const float* in, int nin, const float* __restrict__ W,
                         const float* __restrict__ b, float* out, int nout, bool dorelu)
{
    for (int o = threadIdx.x; o < nout; o += blockDim.x) {
        float a = b[o];
        for (int j = 0; j < nin; ++j) a += in[j] * W[o * nin + j];
        out[o] = dorelu ? fmaxf(a, 0.f) : a;
    }
    __syncthreads();
}

__device__ void softmax_sh(float* v, int n, float* scratch)
{
    if (threadIdx.x == 0) {
        float m = v[0];
        for (int i = 1; i < n; ++i) m = fmaxf(m, v[i]);
        float s = 0.f;
        for (int i = 0; i < n; ++i) { v[i] = expf(v[i] - m); s += v[i]; }
        scratch[0] = 1.0f / s;
    }
    __syncthreads();
    for (int i = threadIdx.x; i < n; i += blockDim.x) v[i] *= scratch[0];
    __syncthreads();
}

__global__ __launch_bounds__(384) void alu_kernel(
    const float* __restrict__ pnrh, int nparts,
    const float* __restrict__ read_head,
    const float* __restrict__ aW1, const float* __restrict__ ab1,
    const float* __restrict__ aW2, const float* __restrict__ ab2,
    const float* __restrict__ aW3, const float* __restrict__ ab3,
    const float* __restrict__ aW4, const float* __restrict__ ab4,
    const float* __restrict__ mW1, const float* __restrict__ mb1,
    const float* __restrict__ mW2, const float* __restrict__ mb2,
    const float* __restrict__ mW3, const float* __restrict__ mb3,
    const float* __restrict__ mW4, const float* __restrict__ mb4,
    const float* __restrict__ Wv, const float* __restrict__ bv,
    float* __restrict__ ctrl, float* __restrict__ out_vec, float* __restrict__ nrh_o)
{
    __shared__ float xin[40], t1[110], t2[190], t3[270], oa[325], om[325], sc[1];
    const int tid = threadIdx.x;
    if (tid < WD) {
        float s = 0.f;
        for (int b = 0; b < nparts; ++b) s += pnrh[b * WD + tid];
        xin[WD + tid] = s;
        nrh_o[tid] = s;
        xin[tid] = read_head[tid];
    }
    __syncthreads();
    // add net
    dense_sh(xin, 40, aW1, ab1, t1, 110, true);
    dense_sh(t1, 110, aW2, ab2, t2, 190, true);
    dense_sh(t2, 190, aW3, ab3, t3, 270, true);
    dense_sh(t3, 270, aW4, ab4, oa, 325, false);
    softmax_sh(oa, 325, sc);
    // mul net
    dense_sh(xin, 40, mW1, mb1, t1, 110, true);
    dense_sh(t1, 110, mW2, mb2, t2, 190, true);
    dense_sh(t2, 190, mW3, mb3, t3, 270, true);
    dense_sh(t3, 270, mW4, mb4, om, 325, false);
    softmax_sh(om, 325, sc);
    // blend + write out
    const float ah0 = ctrl[105], ah1 = ctrl[106];
    for (int o = tid; o < 325; o += blockDim.x) {
        float ov = ah0 * oa[o] + ah1 * om[o];
        oa[o] = ov;
        out_vec[o] = ov;
    }
    __syncthreads();
    // v = out @ Wv.T + bv ; add_final = rho*add_pre + (1-rho)*v
    if (tid < WD) {
        float a = bv[tid];
        for (int c = 0; c < 325; ++c) a += oa[c] * Wv[tid * 325 + c];
        const float rho = ctrl[104];
        ctrl[112 + tid] = rho * ctrl[84 + tid] + (1.0f - rho) * a;
    }
}

// ------------------------- K11: memory update -------------------------
__global__ __launch_bounds__(256) void memupdate_kernel(
    const float* __restrict__ Mem, const float* __restrict__ ww_o,
    const float* __restrict__ ctrl, float* __restrict__ newmem)
{
    __shared__ float er[WD], ad[WD];
    if (threadIdx.x < WD) {
        er[threadIdx.x] = ctrl[64 + threadIdx.x];
        ad[threadIdx.x] = ctrl[112 + threadIdx.x];
    }
    __syncthreads();
    const int total = NROWS * WD;
    const int gid = blockIdx.x * blockDim.x + threadIdx.x;
    const int stride = gridDim.x * blockDim.x;
    for (int e = gid; e < total; e += stride) {
        __builtin_prefetch(Mem + e + stride, 0, 0);   // global_prefetch_b8
        const int i = e / WD;
        const int j = e - i * WD;
        const float wv = ww_o[i];
        const float m = Mem[e];
        newmem[e] = m * (1.0f - wv * er[j]) + wv * ad[j];
    }
}

// ========================= launch =========================
extern "C" void kernel_launch(void* const* d_in, const int* in_sizes, int n_in,
                              void* d_out_v, int out_size, void* d_ws, size_t ws_size,
                              hipStream_t stream)
{
    (void)in_sizes; (void)n_in; (void)out_size; (void)ws_size;
    const float* X    = (const float*)d_in[0];
    const float* rwp  = (const float*)d_in[1];
    const float* wwp  = (const float*)d_in[2];
    const float* Mem  = (const float*)d_in[3];
    const float* rhd  = (const float*)d_in[4];
    const float* W1   = (const float*)d_in[5];
    const float* b1   = (const float*)d_in[6];
    const float* W2   = (const float*)d_in[7];
    const float* b2   = (const float*)d_in[8];
    const float* Wxi  = (const float*)d_in[9];
    const float* bxi  = (const float*)d_in[10];
    const float* Wz   = (const float*)d_in[11];
    const float* bz   = (const float*)d_in[12];
    const float* Wv   = (const float*)d_in[13];
    const float* bv   = (const float*)d_in[14];
    const float* aW1  = (const float*)d_in[15];
    const float* ab1  = (const float*)d_in[16];
    const float* aW2  = (const float*)d_in[17];
    const float* ab2  = (const float*)d_in[18];
    const float* aW3  = (const float*)d_in[19];
    const float* ab3  = (const float*)d_in[20];
    const float* aW4  = (const float*)d_in[21];
    const float* ab4  = (const float*)d_in[22];
    const float* mW1  = (const float*)d_in[23];
    const float* mb1  = (const float*)d_in[24];
    const float* mW2  = (const float*)d_in[25];
    const float* mb2  = (const float*)d_in[26];
    const float* mW3  = (const float*)d_in[27];
    const float* mb3  = (const float*)d_in[28];
    const float* mW4  = (const float*)d_in[29];
    const float* mb4  = (const float*)d_in[30];

    float* out   = (float*)d_out_v;            // 325
    float* rw_o  = out + 325;                  // N
    float* ww_o  = rw_o + NROWS;               // N
    float* nm_o  = ww_o + NROWS;               // 20N (scratch early, final last)
    float* nrh_o = nm_o + (size_t)NROWS * WD;  // 20
    // scratch arrays inside new_memory region (fully overwritten by K11)
    float* sim_r = nm_o;
    float* sim_w = nm_o + NROWS;
    float* wg_r  = nm_o + 2 * (size_t)NROWS;
    float* wg_w  = nm_o + 3 * (size_t)NROWS;

    float* ctrl   = (float*)d_ws;          // 256 floats
    float* pmax_r = ctrl + 256;            // 1024
    float* pmax_w = pmax_r + 1024;
    float* psum_r = pmax_w + 1024;
    float* psum_w = psum_r + 1024;
    float* pws_r  = psum_w + 1024;
    float* pws_w  = pws_r + 1024;
    float* pnrh   = pws_w + 1024;          // 1024 * 20

    const int NB = 1024, TPB = 256;

    controller_kernel<<<1, 128, 0, stream>>>(X, W1, b1, W2, b2, Wxi, bxi, Wz, bz, ctrl);
    score_kernel<<<NB, TPB, 0, stream>>>(Mem, ctrl, sim_r, sim_w, pmax_r, pmax_w);
    reduce_pair_kernel<<<1, 1024, 0, stream>>>(pmax_r, pmax_w, NB, &ctrl[160], &ctrl[161], 0);
    softmax_exp_kernel<<<NB, TPB, 0, stream>>>(sim_r, sim_w, ctrl, psum_r, psum_w);
    reduce_pair_kernel<<<1, 1024, 0, stream>>>(psum_r, psum_w, NB, &ctrl[162], &ctrl[163], 1);
    wg_kernel<<<NB, TPB, 0, stream>>>(sim_r, sim_w, rwp, wwp, ctrl, wg_r, wg_w);
    conv_pow_kernel<<<NB, TPB, 0, stream>>>(wg_r, wg_w, ctrl, rw_o, ww_o, pws_r, pws_w);
    reduce_pair_kernel<<<1, 1024, 0, stream>>>(pws_r, pws_w, NB, &ctrl[164], &ctrl[165], 1);
    norm_kernel<<<2048, TPB, 0, stream>>>(rw_o, ww_o, ctrl);
    readvec_kernel<<<NB, TPB, 0, stream>>>(Mem, rw_o, pnrh);
    alu_kernel<<<1, 384, 0, stream>>>(pnrh, NB, rhd,
        aW1, ab1, aW2, ab2, aW3, ab3, aW4, ab4,
        mW1, mb1, mW2, mb2, mW3, mb3, mW4, mb4,
        Wv, bv, ctrl, out, nrh_o);
    memupdate_kernel<<<2048, TPB, 0, stream>>>(Mem, ww_o, ctrl, nm_o);
}